// Position_Game_75453985456633
// MI455X (gfx1250) — compile-verified
//
#include <hip/hip_runtime.h>
#include <hip/hip_bf16.h>
#include <math.h>

typedef __attribute__((ext_vector_type(2))) float v2f;
typedef __attribute__((ext_vector_type(8))) float v8f;

#define NB   16     // batch
#define NS   32     // subcoalitions
#define NE   32     // hyper-edges
#define NN   128    // nodes
#define ND   128    // state dim
#define HSTR 129    // padded LDS row stride (dwords): bank = (row+n) % 64, conflict-free

// ---------------------------------------------------------------------------
// Kernel 1: w_diag = relu(states@W1 + b1)@W2 + b2 ;  absw = |w_diag|  (16x32)
// One wave32, EXEC all ones (WMMA requirement). Uses V_WMMA_F32_16X16X4_F32.
// A 16x4 layout: lane<16 -> M=lane, K={k0,k0+1}; lane>=16 -> M=lane-16, K={k0+2,k0+3}
// B 4x16 layout: mirrors A's half-wave K split, N = lane&15
// C 16x16 layout: VGPR r -> M = r + 8*(lane>=16), N = lane&15
// ---------------------------------------------------------------------------
__global__ __launch_bounds__(32)
void mlp_wdiag_kernel(const float* __restrict__ states,
                      const float* __restrict__ W1, const float* __restrict__ b1,
                      const float* __restrict__ W2, const float* __restrict__ b2,
                      float* __restrict__ absw) {
  __shared__ float H[NB * ND];
  const int lane = threadIdx.x;
  const int half = lane >> 4;     // 0/1
  const int l15  = lane & 15;
  const int koff = half * 2;

  // GEMM1: H = relu(states(16x128) @ W1(128x128) + b1)
  for (int t = 0; t < 8; ++t) {
    const int nb = t * 16;
    v8f acc = {};
    for (int kb = 0; kb < 32; ++kb) {
      const int kr = kb * 4 + koff;
      v2f a, bm;
      a.x  = states[l15 * ND + kr];
      a.y  = states[l15 * ND + kr + 1];
      bm.x = W1[kr * ND + nb + l15];
      bm.y = W1[(kr + 1) * ND + nb + l15];
      acc = __builtin_amdgcn_wmma_f32_16x16x4_f32(false, a, false, bm,
                                                  (short)0, acc, false, false);
    }
#pragma unroll
    for (int r = 0; r < 8; ++r) {
      const int m = r + half * 8;
      float v = acc[r] + b1[nb + l15];
      H[m * ND + nb + l15] = v > 0.0f ? v : 0.0f;
    }
  }
  __syncthreads();

  // GEMM2: w = H(16x128) @ W2(128x32) + b2 ; store |w|
  for (int t = 0; t < 2; ++t) {
    const int nb = t * 16;
    v8f acc = {};
    for (int kb = 0; kb < 32; ++kb) {
      const int kr = kb * 4 + koff;
      v2f a, bm;
      a.x  = H[l15 * ND + kr];
      a.y  = H[l15 * ND + kr + 1];
      bm.x = W2[kr * NE + nb + l15];
      bm.y = W2[(kr + 1) * NE + nb + l15];
      acc = __builtin_amdgcn_wmma_f32_16x16x4_f32(false, a, false, bm,
                                                  (short)0, acc, false, false);
    }
#pragma unroll
    for (int r = 0; r < 8; ++r) {
      const int m = r + half * 8;
      float v = acc[r] + b2[nb + l15];
      absw[m * NE + nb + l15] = fabsf(v);
    }
  }
}

// ---------------------------------------------------------------------------
// Kernel 2: one block per (b,s). LDS-resident hgT tile (padded stride 129,
// zeroed sentinel row 32 for masked-out gathers -> branchless ds_load+v_max),
// then degree normalizations and the 32x128 matvec to produce X[b,s,:,0].
// ---------------------------------------------------------------------------
__global__ __launch_bounds__(128)
void coalition_kernel(const float* __restrict__ nf,     // (B, NN, 1)
                      const float* __restrict__ hg,     // (B, NN, NE)
                      const int*   __restrict__ subc,   // (B, S, NE, NE)
                      const int*   __restrict__ gcoal,  // (B, S, NE, NE)
                      const float* __restrict__ absw,   // (B, NE)
                      float* __restrict__ out) {        // (B, S, NE, 1)
  __shared__ float s_hg[(NE + 1) * HSTR];   // hgT rows 0..31 + zero row 32
  __shared__ float s_nh[NE * HSTR];         // new_hg [i][n]
  __shared__ int   s_idx[NE * NE];          // gather row (or 32 if masked)
  __shared__ float s_aw[NE];
  __shared__ float s_u[NN];

  const int bs  = blockIdx.x;
  const int b   = bs >> 5;                  // / S
  const int tid = threadIdx.x;

  // hgT tile: coalesced global read (b shared by 32 blocks -> L2 hit),
  // transposed LDS write; padded stride keeps writes conflict-free.
  for (int i = tid; i < NN * NE; i += 128) {
    const int n = i >> 5;
    const int e = i & 31;
    s_hg[e * HSTR + n] = hg[b * (NN * NE) + i];
  }
  s_hg[NE * HSTR + tid] = 0.0f;             // sentinel zero row (n = tid)

  const int base = bs * (NE * NE);
  for (int i = tid; i < NE * NE; i += 128) {
    const int m = subc[base + i];
    s_idx[i] = m ? gcoal[base + i] : NE;    // NE -> zero row
  }
  if (tid < NE) s_aw[tid] = absw[b * NE + tid];
  __syncthreads();

  // new_hg[i][n] = max_j (mask * hgT[gc][n]); start at -inf like jnp.max
  const int n = tid;                        // 0..127, column owner
  for (int i = 0; i < NE; ++i) {
    float m = -INFINITY;
#pragma unroll
    for (int j4 = 0; j4 < 8; ++j4) {
      int4 q = *(const int4*)&s_idx[i * NE + j4 * 4];   // ds_load_b128, broadcast
      float v0 = s_hg[q.x * HSTR + n];
      float v1 = s_hg[q.y * HSTR + n];
      float v2 = s_hg[q.z * HSTR + n];
      float v3 = s_hg[q.w * HSTR + n];
      m = fmaxf(m, fmaxf(fmaxf(v0, v1), fmaxf(v2, v3)));
    }
    s_nh[i * HSTR + n] = m;
  }
  __syncthreads();

  // d_deg[n] = sum_i new_hg[i][n]*|w|[i]; u[n] = d_inv[n]*nf[n]
  float dd = 0.0f;
  for (int i = 0; i < NE; ++i) dd += s_nh[i * HSTR + n] * s_aw[i];
  const float dinv = (dd > 0.0f) ? rsqrtf(dd) : 0.0f;
  s_u[n] = dinv * nf[b * NN + n];
  __syncthreads();

  // b_deg[i] and X[i] = |w|[i]*b_inv[i]*sum_n u[n]*new_hg[i][n]
  if (tid < NE) {
    const int i = tid;
    float bd = 0.0f, acc = 0.0f;
    for (int nn = 0; nn < NN; ++nn) {
      const float v = s_nh[i * HSTR + nn];
      bd  += v;
      acc += s_u[nn] * v;
    }
    const float binv = (bd != 0.0f) ? (1.0f / bd) : 0.0f;
    out[bs * NE + i] = s_aw[i] * binv * acc;
  }
}

extern "C" void kernel_launch(void* const* d_in, const int* in_sizes, int n_in,
                              void* d_out, int out_size, void* d_ws, size_t ws_size,
                              hipStream_t stream) {
  const float* node_features = (const float*)d_in[0];
  const float* hyper_graph   = (const float*)d_in[1];
  const float* states        = (const float*)d_in[2];
  const int*   subc          = (const int*)  d_in[3];
  const int*   gcoal         = (const int*)  d_in[4];
  const float* W1            = (const float*)d_in[5];
  const float* b1            = (const float*)d_in[6];
  const float* W2            = (const float*)d_in[7];
  const float* b2            = (const float*)d_in[8];
  float* out  = (float*)d_out;
  float* absw = (float*)d_ws;               // 16*32 floats

  mlp_wdiag_kernel<<<1, 32, 0, stream>>>(states, W1, b1, W2, b2, absw);
  coalition_kernel<<<NB * NS, 128, 0, stream>>>(node_features, hyper_graph,
                                                subc, gcoal, absw, out);
}